// BaselineRvNNModel_43121471652508
// MI455X (gfx1250) — compile-verified
//
#include <hip/hip_runtime.h>

typedef float v2f __attribute__((ext_vector_type(2)));
typedef float v8f __attribute__((ext_vector_type(8)));

#define D_IN   768
#define H_DIM  256
#define ROWS   48      // rows per workgroup
#define XS     68      // x stage LDS stride (floats), 68 % 64 == 4 -> conflict-free
#define HS     260     // h LDS stride (floats), 260 % 64 == 4 -> conflict-free
#define LN_EPS 1e-5f

__device__ __forceinline__ float sigm(float v) { return 1.0f / (1.0f + __expf(-v)); }

__global__ void init_pool_kernel(float* __restrict__ pool) {
    pool[threadIdx.x] = 0.0f;   // 256 threads, one block
}

__global__ __launch_bounds__(256)
void fused_rvnn_kernel(const float* __restrict__ x,
                       const float* __restrict__ W1,  const float* __restrict__ b1,
                       const float* __restrict__ W2,  const float* __restrict__ b2,
                       const float* __restrict__ lnw, const float* __restrict__ lnb,
                       const float* __restrict__ Wiou,
                       const float* __restrict__ bwio, const float* __restrict__ buio,
                       float* __restrict__ pool, int N)
{
    __shared__ float hs[ROWS * HS];   // h1 then h2 (in-place LN)
    __shared__ float xs[ROWS * XS];   // K-chunk stage of x
    __shared__ float csum[H_DIM];     // per-WG pooled column sums

    const int tid    = threadIdx.x;
    const int wave   = tid >> 5;
    const int lane   = tid & 31;
    const int hf     = lane >> 4;     // 0: lanes 0-15, 1: lanes 16-31
    const int lq     = lane & 15;
    const int wgBase = blockIdx.x * ROWS;

    if (tid < H_DIM) csum[tid] = 0.0f;

    // ================= GEMM1: h1 = relu(x @ W1^T + b1) =================
    v8f acc[6];
    #pragma unroll
    for (int i = 0; i < 6; ++i) acc[i] = (v8f){};

    for (int kc = 0; kc < D_IN / 64; ++kc) {
        __syncthreads();
        // stage x[:, kc*64 .. kc*64+64) -> xs (zero-fill tail rows)
        for (int idx = tid; idx < ROWS * 16; idx += 256) {
            int r = idx >> 4, q = idx & 15;
            int grow = wgBase + r;
            float4 v = make_float4(0.f, 0.f, 0.f, 0.f);
            if (grow < N)
                v = *(const float4*)(x + (size_t)grow * D_IN + kc * 64 + q * 4);
            *(float4*)(xs + r * XS + q * 4) = v;
        }
        __syncthreads();

        #pragma unroll
        for (int i = 0; i < 6; ++i) {
            int t  = wave + 8 * i;          // tile id 0..47
            int rt = t >> 4, cc = t & 15;
            const float* arow = xs + (rt * 16 + lq) * XS + 2 * hf;
            const float* brow = W1 + (size_t)(cc * 16 + lq) * D_IN + kc * 64 + 2 * hf;
            #pragma unroll
            for (int k4 = 0; k4 < 16; ++k4) {
                v2f a = *(const v2f*)(arow + k4 * 4);
                v2f b = *(const v2f*)(brow + k4 * 4);
                acc[i] = __builtin_amdgcn_wmma_f32_16x16x4_f32(
                    false, a, false, b, (short)0, acc[i], false, false);
            }
        }
    }
    __syncthreads();
    // bias + relu -> hs
    #pragma unroll
    for (int i = 0; i < 6; ++i) {
        int t  = wave + 8 * i;
        int rt = t >> 4, cc = t & 15;
        float bias = b1[cc * 16 + lq];
        #pragma unroll
        for (int v = 0; v < 8; ++v) {
            int m = v + 8 * hf;
            float val = acc[i][v] + bias;
            hs[(rt * 16 + m) * HS + cc * 16 + lq] = val > 0.f ? val : 0.f;
        }
    }
    __syncthreads();

    // ================= GEMM2: h2 = h1 @ W2^T + b2 (then LN) =================
    v8f acc2[6];
    #pragma unroll
    for (int i = 0; i < 6; ++i) acc2[i] = (v8f){};

    #pragma unroll
    for (int i = 0; i < 6; ++i) {
        int t  = wave + 8 * i;
        int rt = t >> 4, cc = t & 15;
        const float* arow = hs + (rt * 16 + lq) * HS + 2 * hf;
        const float* brow = W2 + (size_t)(cc * 16 + lq) * H_DIM + 2 * hf;
        #pragma unroll 8
        for (int k4 = 0; k4 < 64; ++k4) {
            v2f a = *(const v2f*)(arow + k4 * 4);
            v2f b = *(const v2f*)(brow + k4 * 4);
            acc2[i] = __builtin_amdgcn_wmma_f32_16x16x4_f32(
                false, a, false, b, (short)0, acc2[i], false, false);
        }
    }
    __syncthreads();   // all reads of h1 complete; safe to overwrite hs
    #pragma unroll
    for (int i = 0; i < 6; ++i) {
        int t  = wave + 8 * i;
        int rt = t >> 4, cc = t & 15;
        float bias = b2[cc * 16 + lq];
        #pragma unroll
        for (int v = 0; v < 8; ++v) {
            int m = v + 8 * hf;
            hs[(rt * 16 + m) * HS + cc * 16 + lq] = acc2[i][v] + bias;
        }
    }
    __syncthreads();
    // LayerNorm rows in place (one thread per row)
    if (tid < ROWS) {
        float* row = hs + tid * HS;
        float s = 0.f;
        for (int j = 0; j < H_DIM; ++j) s += row[j];
        float mu = s * (1.0f / H_DIM);
        float var = 0.f;
        for (int j = 0; j < H_DIM; ++j) { float d = row[j] - mu; var += d * d; }
        float inv = rsqrtf(var * (1.0f / H_DIM) + LN_EPS);
        for (int j = 0; j < H_DIM; ++j)
            row[j] = (row[j] - mu) * inv * lnw[j] + lnb[j];
    }
    __syncthreads();

    // ===== GEMM3: iou = h2 @ W_iou^T + b; gates + pooled column sums =====
    for (int i = 0; i < 6; ++i) {
        int t  = wave + 8 * i;
        int rt = t >> 4, cc = t & 15;
        v8f ai = (v8f){}, ao = (v8f){}, au = (v8f){};
        const float* arow = hs + (rt * 16 + lq) * HS + 2 * hf;
        const float* bi_ = Wiou + (size_t)(cc * 16 + lq) * H_DIM + 2 * hf;
        const float* bo_ = Wiou + (size_t)(H_DIM + cc * 16 + lq) * H_DIM + 2 * hf;
        const float* bu_ = Wiou + (size_t)(2 * H_DIM + cc * 16 + lq) * H_DIM + 2 * hf;
        #pragma unroll 8
        for (int k4 = 0; k4 < 64; ++k4) {
            v2f a = *(const v2f*)(arow + k4 * 4);
            ai = __builtin_amdgcn_wmma_f32_16x16x4_f32(
                false, a, false, *(const v2f*)(bi_ + k4 * 4), (short)0, ai, false, false);
            ao = __builtin_amdgcn_wmma_f32_16x16x4_f32(
                false, a, false, *(const v2f*)(bo_ + k4 * 4), (short)0, ao, false, false);
            au = __builtin_amdgcn_wmma_f32_16x16x4_f32(
                false, a, false, *(const v2f*)(bu_ + k4 * 4), (short)0, au, false, false);
        }
        int n = cc * 16 + lq;
        float Bi = bwio[n]             + buio[n];
        float Bo = bwio[H_DIM + n]     + buio[H_DIM + n];
        float Bu = bwio[2 * H_DIM + n] + buio[2 * H_DIM + n];
        float s = 0.f;
        #pragma unroll
        for (int v = 0; v < 8; ++v) {
            int m = v + 8 * hf;
            int grow = wgBase + rt * 16 + m;
            if (grow < N) {
                float ig = sigm(ai[v] + Bi);
                float ug = tanhf(au[v] + Bu);
                float og = sigm(ao[v] + Bo);
                s += og * tanhf(ig * ug);       // hn = sig(o)*tanh(sig(i)*tanh(u))
            }
        }
        s += __shfl_down(s, 16, 32);            // combine m-halves (same column n)
        if (hf == 0) atomicAdd(&csum[n], s);
    }
    __syncthreads();
    if (tid < H_DIM) atomicAdd(&pool[tid], csum[tid]);
}

__global__ __launch_bounds__(256)
void finalize_kernel(const float* __restrict__ pool,
                     const float* __restrict__ Wc1, const float* __restrict__ bc1,
                     const float* __restrict__ Wc2, const float* __restrict__ bc2,
                     float* __restrict__ out, int N)
{
    __shared__ float p[H_DIM];
    __shared__ float z[H_DIM / 2];
    int tid = threadIdx.x;
    p[tid] = pool[tid] / (float)N;            // global mean pool
    __syncthreads();
    if (tid < H_DIM / 2) {
        float s = bc1[tid];
        for (int k = 0; k < H_DIM; ++k) s += p[k] * Wc1[tid * H_DIM + k];
        z[tid] = s > 0.f ? s : 0.f;
    }
    __syncthreads();
    if (tid < 4) {
        float s = bc2[tid];
        for (int k = 0; k < H_DIM / 2; ++k) s += z[k] * Wc2[tid * (H_DIM / 2) + k];
        out[tid] = s;
    }
}

extern "C" void kernel_launch(void* const* d_in, const int* in_sizes, int n_in,
                              void* d_out, int out_size, void* d_ws, size_t ws_size,
                              hipStream_t stream)
{
    const float* x    = (const float*)d_in[0];
    /* d_in[1] = edge_index (unused: tree collapses to per-node gates) */
    const float* W1   = (const float*)d_in[2];
    const float* b1   = (const float*)d_in[3];
    const float* W2   = (const float*)d_in[4];
    const float* b2   = (const float*)d_in[5];
    const float* lnw  = (const float*)d_in[6];
    const float* lnb  = (const float*)d_in[7];
    const float* Wiou = (const float*)d_in[8];
    const float* bwio = (const float*)d_in[9];
    const float* buio = (const float*)d_in[10];
    const float* Wc1  = (const float*)d_in[11];
    const float* bc1  = (const float*)d_in[12];
    const float* Wc2  = (const float*)d_in[13];
    const float* bc2  = (const float*)d_in[14];
    float* out  = (float*)d_out;
    float* pool = (float*)d_ws;               // 256 floats of scratch

    int N   = in_sizes[0] / D_IN;
    int nwg = (N + ROWS - 1) / ROWS;

    hipLaunchKernelGGL(init_pool_kernel, dim3(1), dim3(256), 0, stream, pool);
    hipLaunchKernelGGL(fused_rvnn_kernel, dim3(nwg), dim3(256), 0, stream,
                       x, W1, b1, W2, b2, lnw, lnb, Wiou, bwio, buio, pool, N);
    hipLaunchKernelGGL(finalize_kernel, dim3(1), dim3(256), 0, stream,
                       pool, Wc1, bc1, Wc2, bc2, out, N);
}